// LSTMAttentionDecoder01_74148315398635
// MI455X (gfx1250) — compile-verified
//
#include <hip/hip_runtime.h>
#include <math.h>

typedef __attribute__((ext_vector_type(2))) float v2f;
typedef __attribute__((ext_vector_type(8))) float v8f;

#define TY   50
#define BATCH 32
#define TX   400
#define HDIM 512
#define CTXD 1024
#define DYD  512
#define VOC  50000
#define EXTN 50

__device__ __forceinline__ float sigf(float x) { return 1.0f / (1.0f + expf(-x)); }

#define WMMA_F32(a, b, c) \
    __builtin_amdgcn_wmma_f32_16x16x4_f32(false, (a), false, (b), (short)0, (c), false, false)

// ---------------------------------------------------------------------------
// Generic dual-input WMMA GEMM:  C[M,N] = A[M,K1] @ W1[N,K1]^T
//                                       (+ A2[M,K2] @ W2[N,K2]^T)
//                                       (+ bias1[N] + bias2[N]) , optional tanh
// fp32 via V_WMMA_F32_16X16X4_F32.  4 waves/block.  Each wave computes a
// 32-row x 64-col block (2 M-tiles x 4 N-tiles): the 2 M-tiles share every B
// fragment, halving weight traffic (the dominant L2 stream for skinny-M
// GEMMs).  Branch-free inner loop; N-tail handled by clamping the strip start
// (overlapping waves recompute identical values -> deterministic duplicate
// stores).  Requires: K % 8 == 0, M % 32 == 0, N >= 64, N % 16 == 0.
// grid = ((N+255)/256, M/32), block = 128.
// ---------------------------------------------------------------------------
__global__ __launch_bounds__(128) void wmma_gemm(
    const float* __restrict__ A,  int lda,  int K1, const float* __restrict__ W1,
    const float* __restrict__ A2, int lda2, int K2, const float* __restrict__ W2,
    const float* __restrict__ bias1, const float* __restrict__ bias2,
    float* __restrict__ C, int ldc, int N, int act)
{
    const int lane = threadIdx.x & 31;
    const int wave = threadIdx.x >> 5;
    const int nl   = lane & 15;      // col / row-in-tile index
    const int half = lane >> 4;      // K sub-pair selector
    const int koff = half * 2;
    const int mbase = blockIdx.y * 32;                 // first row of block
    int nbase = blockIdx.x * 256 + wave * 64;          // first column of strip
    if (nbase + 64 > N) nbase = N - 64;                // clamp tail (N >= 64)

    v8f acc[2][4] = {};

    auto run = [&](const float* Ap, int ldA, int K, const float* Wp) {
        const float* ap0 = Ap + (size_t)(mbase +  0 + nl) * ldA + koff;
        const float* ap1 = Ap + (size_t)(mbase + 16 + nl) * ldA + koff;
        const float* wp0 = Wp + (size_t)(nbase +  0 + nl) * K + koff;
        const float* wp1 = Wp + (size_t)(nbase + 16 + nl) * K + koff;
        const float* wp2 = Wp + (size_t)(nbase + 32 + nl) * K + koff;
        const float* wp3 = Wp + (size_t)(nbase + 48 + nl) * K + koff;
        for (int k = 0; k < K; k += 8) {
            v2f a00 = *(const v2f*)(ap0 + k);
            v2f a01 = *(const v2f*)(ap0 + k + 4);
            v2f a10 = *(const v2f*)(ap1 + k);
            v2f a11 = *(const v2f*)(ap1 + k + 4);
            v2f b00 = *(const v2f*)(wp0 + k);
            v2f b10 = *(const v2f*)(wp1 + k);
            v2f b20 = *(const v2f*)(wp2 + k);
            v2f b30 = *(const v2f*)(wp3 + k);
            v2f b01 = *(const v2f*)(wp0 + k + 4);
            v2f b11 = *(const v2f*)(wp1 + k + 4);
            v2f b21 = *(const v2f*)(wp2 + k + 4);
            v2f b31 = *(const v2f*)(wp3 + k + 4);
            acc[0][0] = WMMA_F32(a00, b00, acc[0][0]);
            acc[0][1] = WMMA_F32(a00, b10, acc[0][1]);
            acc[0][2] = WMMA_F32(a00, b20, acc[0][2]);
            acc[0][3] = WMMA_F32(a00, b30, acc[0][3]);
            acc[1][0] = WMMA_F32(a10, b00, acc[1][0]);
            acc[1][1] = WMMA_F32(a10, b10, acc[1][1]);
            acc[1][2] = WMMA_F32(a10, b20, acc[1][2]);
            acc[1][3] = WMMA_F32(a10, b30, acc[1][3]);
            acc[0][0] = WMMA_F32(a01, b01, acc[0][0]);
            acc[0][1] = WMMA_F32(a01, b11, acc[0][1]);
            acc[0][2] = WMMA_F32(a01, b21, acc[0][2]);
            acc[0][3] = WMMA_F32(a01, b31, acc[0][3]);
            acc[1][0] = WMMA_F32(a11, b01, acc[1][0]);
            acc[1][1] = WMMA_F32(a11, b11, acc[1][1]);
            acc[1][2] = WMMA_F32(a11, b21, acc[1][2]);
            acc[1][3] = WMMA_F32(a11, b31, acc[1][3]);
        }
    };

    run(A, lda, K1, W1);
    if (A2) run(A2, lda2, K2, W2);

#pragma unroll
    for (int mt = 0; mt < 2; ++mt) {
#pragma unroll
        for (int i = 0; i < 4; ++i) {
            int n = nbase + i * 16 + nl;
            float bs = 0.0f;
            if (bias1) bs += bias1[n];
            if (bias2) bs += bias2[n];
#pragma unroll
            for (int r = 0; r < 8; ++r) {
                int mm = mbase + mt * 16 + r + half * 8;  // C/D: M = r + 8*half
                float vv = acc[mt][i][r] + bs;
                if (act) vv = tanhf(vv);
                C[(size_t)mm * ldc + n] = vv;
            }
        }
    }
}

// ---------------------------------------------------------------------------
// init: copy h0,c0,init_coverage into scratch state
// ---------------------------------------------------------------------------
__global__ void init_state(const float* __restrict__ h0, const float* __restrict__ c0,
                           const float* __restrict__ cov,
                           float* __restrict__ h, float* __restrict__ c,
                           float* __restrict__ acc)
{
    int i = blockIdx.x * 256 + threadIdx.x;
    if (i < BATCH * HDIM) { h[i] = h0[i]; c[i] = c0[i]; }
    if (i < BATCH * TX)   acc[i] = cov[i];
}

// ---------------------------------------------------------------------------
// LSTM cell with parametrized gate ordering + y_mask blend.
// Optionally writes s=[h|c] (32,1024) and mirrored copies of h/c.
// ---------------------------------------------------------------------------
__global__ __launch_bounds__(256) void lstm_cell(
    const float* __restrict__ gates, const float* __restrict__ hprev,
    const float* __restrict__ cprev, const float* __restrict__ ym,
    float* __restrict__ hout, float* __restrict__ cout,
    float* __restrict__ sout, float* __restrict__ hout2, float* __restrict__ cout2,
    int oi, int of, int oc, int oo)
{
    int idx = blockIdx.x * 256 + threadIdx.x;
    if (idx >= BATCH * HDIM) return;
    int b = idx >> 9, j = idx & 511;
    const float* gr = gates + (size_t)b * 4 * HDIM;
    float gi = gr[oi * HDIM + j], gf = gr[of * HDIM + j];
    float gc = gr[oc * HDIM + j], go = gr[oo * HDIM + j];
    float m  = ym[b];
    float cp = cprev[idx], hp = hprev[idx];
    float cn = sigf(gf) * cp + sigf(gi) * tanhf(gc);
    float hn = sigf(go) * tanhf(cn);
    hn = m * hn + (1.0f - m) * hp;
    cn = m * cn + (1.0f - m) * cp;
    hout[idx] = hn; cout[idx] = cn;
    if (sout)  { sout[b * 1024 + j] = hn; sout[b * 1024 + 512 + j] = cn; }
    if (hout2) hout2[idx] = hn;
    if (cout2) cout2[idx] = cn;
}

// ---------------------------------------------------------------------------
// Attention scores: scores[t,b] = xm * sum_c tanh(pctx+hq+acc*wcov)*U
// ---------------------------------------------------------------------------
__global__ __launch_bounds__(256) void att_score(
    const float* __restrict__ pctx, const float* __restrict__ hq,
    const float* __restrict__ acc,  const float* __restrict__ wcov,
    const float* __restrict__ uatt, const float* __restrict__ xmask,
    float* __restrict__ scores)
{
    int t = blockIdx.x, b = blockIdx.y;
    const float* p = pctx + ((size_t)t * BATCH + b) * CTXD;
    const float* h = hq + (size_t)b * CTXD;
    float a = acc[b * TX + t];
    float s = 0.0f;
    for (int c = threadIdx.x; c < CTXD; c += 256)
        s += tanhf(p[c] + h[c] + a * wcov[c]) * uatt[c];
    __shared__ float red[256];
    red[threadIdx.x] = s; __syncthreads();
    for (int off = 128; off > 0; off >>= 1) {
        if (threadIdx.x < off) red[threadIdx.x] += red[threadIdx.x + off];
        __syncthreads();
    }
    if (threadIdx.x == 0) scores[t * BATCH + b] = red[0] * xmask[t * BATCH + b];
}

// ---------------------------------------------------------------------------
// Softmax over Tx, write dists/Cs, update coverage, compute atted[b,:]
// One block per batch row.
// ---------------------------------------------------------------------------
__global__ __launch_bounds__(256) void att_softmax(
    const float* __restrict__ scores, const float* __restrict__ xmask,
    const float* __restrict__ context, float* __restrict__ acc,
    float* __restrict__ dists_sl, float* __restrict__ cs_sl,
    float* __restrict__ atts_sl)
{
    int b = blockIdx.x, tid = threadIdx.x;
    __shared__ float w[TX];
    __shared__ float red[256];

    float lm = -3.4e38f;
    for (int t = tid; t < TX; t += 256) lm = fmaxf(lm, scores[t * BATCH + b]);
    red[tid] = lm; __syncthreads();
    for (int off = 128; off > 0; off >>= 1) {
        if (tid < off) red[tid] = fmaxf(red[tid], red[tid + off]);
        __syncthreads();
    }
    float mx = red[0]; __syncthreads();

    float ls = 0.0f;
    for (int t = tid; t < TX; t += 256) {
        float e = expf(scores[t * BATCH + b] - mx) * xmask[t * BATCH + b];
        w[t] = e; ls += e;
    }
    red[tid] = ls; __syncthreads();
    for (int off = 128; off > 0; off >>= 1) {
        if (tid < off) red[tid] += red[tid + off];
        __syncthreads();
    }
    float inv = 1.0f / red[0]; __syncthreads();

    for (int t = tid; t < TX; t += 256) {
        float ww = w[t] * inv; w[t] = ww;
        dists_sl[b * TX + t] = ww;               // watt[b,t]
        float old = acc[b * TX + t];
        cs_sl[b * TX + t] = old;                 // Cs = pre-update coverage
        acc[b * TX + t] = old + ww;
    }
    __syncthreads();

    for (int c = tid; c < CTXD; c += 256) {
        float s = 0.0f;
        for (int t = 0; t < TX; ++t)
            s += w[t] * context[((size_t)t * BATCH + b) * CTXD + c];
        atts_sl[b * CTXD + c] = s;
    }
}

// ---------------------------------------------------------------------------
// hcat = [h2 | atted | x_t]  (32, 2048)
// ---------------------------------------------------------------------------
__global__ void build_hcat(const float* __restrict__ h2, const float* __restrict__ atted,
                           const float* __restrict__ xt, float* __restrict__ hcat)
{
    int i = blockIdx.x * 256 + threadIdx.x;
    if (i >= BATCH * 2048) return;
    int b = i >> 11, j = i & 2047;
    float vv;
    if (j < 512)        vv = h2[b * 512 + j];
    else if (j < 1536)  vv = atted[b * 1024 + (j - 512)];
    else                vv = xt[b * 512 + (j - 1536)];
    hcat[i] = vv;
}

// ---------------------------------------------------------------------------
// g[b] = sigmoid(hcat[b,:] . v + bv)
// ---------------------------------------------------------------------------
__global__ __launch_bounds__(256) void gate_g(
    const float* __restrict__ hcat, const float* __restrict__ vw,
    const float* __restrict__ bv, float* __restrict__ g)
{
    int b = blockIdx.x;
    float s = 0.0f;
    for (int j = threadIdx.x; j < 2048; j += 256) s += hcat[b * 2048 + j] * vw[j];
    __shared__ float red[256];
    red[threadIdx.x] = s; __syncthreads();
    for (int off = 128; off > 0; off >>= 1) {
        if (threadIdx.x < off) red[threadIdx.x] += red[threadIdx.x + off];
        __syncthreads();
    }
    if (threadIdx.x == 0) g[b] = sigf(red[0] + bv[0]);
}

// ---------------------------------------------------------------------------
// Vocab softmax + copy-gate mix + pointer scatter-add into ydec slice.
// One block per batch row; thread 0 serializes the 400 scatter-adds (same row,
// duplicate indices possible).  Non-temporal stores for the huge write-once
// ydec stream so it doesn't evict the L2-resident w_logit.
// ---------------------------------------------------------------------------
__global__ __launch_bounds__(512) void ydec_softmax(
    const float* __restrict__ logit, const float* __restrict__ g,
    const float* __restrict__ watt, const int* __restrict__ xid,
    float* __restrict__ y)
{
    int b = blockIdx.x, tid = threadIdx.x;
    const float* lr = logit + (size_t)b * VOC;
    __shared__ float red[512];

    float lm = -3.4e38f;
    for (int v2 = tid; v2 < VOC; v2 += 512) lm = fmaxf(lm, lr[v2]);
    red[tid] = lm; __syncthreads();
    for (int off = 256; off > 0; off >>= 1) {
        if (tid < off) red[tid] = fmaxf(red[tid], red[tid + off]);
        __syncthreads();
    }
    float mx = red[0]; __syncthreads();

    float ls = 0.0f;
    for (int v2 = tid; v2 < VOC; v2 += 512) ls += expf(lr[v2] - mx);
    red[tid] = ls; __syncthreads();
    for (int off = 256; off > 0; off >>= 1) {
        if (tid < off) red[tid] += red[tid + off];
        __syncthreads();
    }
    float sm = red[0];

    float gb = g[b];
    float scale = gb / sm;
    float* yr = y + (size_t)b * (VOC + EXTN);
    for (int v2 = tid; v2 < VOC; v2 += 512)
        __builtin_nontemporal_store(expf(lr[v2] - mx) * scale, yr + v2);
    for (int v2 = VOC + tid; v2 < VOC + EXTN; v2 += 512)
        __builtin_nontemporal_store(0.0f, yr + v2);
    __syncthreads();

    if (tid == 0) {
        float og = 1.0f - gb;
        for (int t = 0; t < TX; ++t) {
            int id = xid[t * BATCH + b];
            yr[id] += og * watt[b * TX + t];
        }
    }
}

// ---------------------------------------------------------------------------
extern "C" void kernel_launch(void* const* d_in, const int* in_sizes, int n_in,
                              void* d_out, int out_size, void* d_ws, size_t ws_size,
                              hipStream_t stream)
{
    const float* y_emb      = (const float*)d_in[0];
    const float* context    = (const float*)d_in[1];
    const float* h0         = (const float*)d_in[2];
    const float* c0         = (const float*)d_in[3];
    const float* x_mask     = (const float*)d_in[4];
    const float* y_mask     = (const float*)d_in[5];
    const float* init_cov   = (const float*)d_in[6];
    const float* Wih        = (const float*)d_in[7];
    const float* Whh        = (const float*)d_in[8];
    const float* bih        = (const float*)d_in[9];
    const float* bhh        = (const float*)d_in[10];
    const float* Wx         = (const float*)d_in[11];
    const float* Ux         = (const float*)d_in[12];
    const float* bx         = (const float*)d_in[13];
    const float* Wc_att     = (const float*)d_in[14];
    const float* b_att      = (const float*)d_in[15];
    const float* W_comb_att = (const float*)d_in[16];
    const float* U_att      = (const float*)d_in[17];
    const float* W_cov      = (const float*)d_in[18];
    const float* w_ds       = (const float*)d_in[19];
    const float* b_ds       = (const float*)d_in[20];
    const float* w_logit    = (const float*)d_in[21];
    const float* b_logit    = (const float*)d_in[22];
    const float* vw         = (const float*)d_in[23];
    const float* bv         = (const float*)d_in[24];
    const int*   xid        = (const int*)d_in[25];

    // scratch carve-out
    float* ws = (float*)d_ws;
    size_t off = 0;
    auto carve = [&](size_t n) { float* p = ws + off; off += n; return p; };
    float* pctx   = carve((size_t)TX * BATCH * CTXD);   // 13.1M
    float* gates  = carve((size_t)BATCH * 4 * HDIM);
    float* pre    = carve((size_t)BATCH * 4 * HDIM);
    float* hq     = carve((size_t)BATCH * CTXD);
    float* scores = carve((size_t)TX * BATCH);
    float* h_scr  = carve((size_t)BATCH * HDIM);
    float* c_scr  = carve((size_t)BATCH * HDIM);
    float* h1b    = carve((size_t)BATCH * HDIM);
    float* c1b    = carve((size_t)BATCH * HDIM);
    float* accb   = carve((size_t)BATCH * TX);
    float* hcat   = carve((size_t)BATCH * 2048);
    float* dstate = carve((size_t)BATCH * HDIM);
    float* gbuf   = carve((size_t)BATCH);
    float* logit  = carve((size_t)BATCH * VOC);         // 1.6M

    // output carve-out: ((hs, cs), ss, atts, dists, Cs, ydecs)
    float* hs    = (float*)d_out;
    float* cs    = hs    + (size_t)TY * BATCH * HDIM;
    float* ss    = cs    + (size_t)TY * BATCH * HDIM;
    float* atts  = ss    + (size_t)TY * BATCH * 2 * HDIM;
    float* dists = atts  + (size_t)TY * BATCH * CTXD;
    float* Cso   = dists + (size_t)TY * BATCH * TX;
    float* ydec  = Cso   + (size_t)TY * BATCH * TX;

    init_state<<<(BATCH * HDIM + 255) / 256, 256, 0, stream>>>(h0, c0, init_cov,
                                                               h_scr, c_scr, accb);

    // pctx = context @ Wc_att^T + b_att   (M=12800, N=1024, K=1024)
    wmma_gemm<<<dim3((CTXD + 255) / 256, (TX * BATCH) / 32), 128, 0, stream>>>(
        context, CTXD, CTXD, Wc_att, nullptr, 0, 0, nullptr,
        b_att, nullptr, pctx, CTXD, CTXD, 0);

    for (int t = 0; t < TY; ++t) {
        const float* xt = y_emb + (size_t)t * BATCH * DYD;
        const float* ym = y_mask + (size_t)t * BATCH;
        float* hs_sl = hs    + (size_t)t * BATCH * HDIM;
        float* cs_sl = cs    + (size_t)t * BATCH * HDIM;
        float* ss_sl = ss    + (size_t)t * BATCH * 2 * HDIM;
        float* at_sl = atts  + (size_t)t * BATCH * CTXD;
        float* di_sl = dists + (size_t)t * BATCH * TX;
        float* Cs_sl = Cso   + (size_t)t * BATCH * TX;
        float* yd_sl = ydec  + (size_t)t * BATCH * (VOC + EXTN);

        // gates = x_t@Wih^T + h_prev@Whh^T + bih + bhh   (N=2048)
        wmma_gemm<<<dim3((4 * HDIM + 255) / 256, BATCH / 32), 128, 0, stream>>>(
            xt, DYD, DYD, Wih, h_scr, HDIM, HDIM, Whh,
            bih, bhh, gates, 4 * HDIM, 4 * HDIM, 0);

        // cell 1 (order i,f,g,o); writes h1,c1 and s=[h1|c1] into ss slice
        lstm_cell<<<(BATCH * HDIM + 255) / 256, 256, 0, stream>>>(
            gates, h_scr, c_scr, ym, h1b, c1b, ss_sl, nullptr, nullptr, 0, 1, 2, 3);

        // hq = s @ W_comb_att^T   (N=1024, K=1024)
        wmma_gemm<<<dim3((CTXD + 255) / 256, BATCH / 32), 128, 0, stream>>>(
            ss_sl, 2 * HDIM, 2 * HDIM, W_comb_att, nullptr, 0, 0, nullptr,
            nullptr, nullptr, hq, CTXD, CTXD, 0);

        att_score<<<dim3(TX, BATCH), 256, 0, stream>>>(
            pctx, hq, accb, W_cov, U_att, x_mask, scores);

        att_softmax<<<BATCH, 256, 0, stream>>>(
            scores, x_mask, context, accb, di_sl, Cs_sl, at_sl);

        // pre = h1@Ux^T + atted@Wx^T + bx   (N=2048)
        wmma_gemm<<<dim3((4 * HDIM + 255) / 256, BATCH / 32), 128, 0, stream>>>(
            h1b, HDIM, HDIM, Ux, at_sl, CTXD, CTXD, Wx,
            bx, nullptr, pre, 4 * HDIM, 4 * HDIM, 0);

        // cell 2 (order i,f,o,c); prev for blend is h1/c1; writes h2/c2 to
        // state scratch and to hs/cs output slices
        lstm_cell<<<(BATCH * HDIM + 255) / 256, 256, 0, stream>>>(
            pre, h1b, c1b, ym, h_scr, c_scr, nullptr, hs_sl, cs_sl, 0, 1, 3, 2);

        build_hcat<<<(BATCH * 2048 + 255) / 256, 256, 0, stream>>>(
            h_scr, at_sl, xt, hcat);

        // dstate = tanh(hcat @ w_ds^T + b_ds)   (N=512, K=2048, fused tanh)
        wmma_gemm<<<dim3((HDIM + 255) / 256, BATCH / 32), 128, 0, stream>>>(
            hcat, 2048, 2048, w_ds, nullptr, 0, 0, nullptr,
            b_ds, nullptr, dstate, HDIM, HDIM, 1);

        gate_g<<<BATCH, 256, 0, stream>>>(hcat, vw, bv, gbuf);

        // logit = dstate @ w_logit^T + b_logit  (N=50000, K=512; L2-resident W,
        // now read exactly once per step thanks to 32-row waves)
        wmma_gemm<<<dim3((VOC + 255) / 256, BATCH / 32), 128, 0, stream>>>(
            dstate, HDIM, HDIM, w_logit, nullptr, 0, 0, nullptr,
            b_logit, nullptr, logit, VOC, VOC, 0);

        ydec_softmax<<<BATCH, 512, 0, stream>>>(logit, gbuf, di_sl, xid, yd_sl);
    }
}